// MaskIoUPostProcessor_56238301773972
// MI455X (gfx1250) — compile-verified
//
#include <hip/hip_runtime.h>
#include <hip/hip_bf16.h>

#define NBOX 2048
#define NCLS 81
#define NIMG 8
#define THREADS 1024
#define SCORE_THRESH 0.05f
#define NMS_THRESH 0.5f
#define DET_PER_IMG 100
#define CLASS_OFFSET 10000.0f

// ---------------------------------------------------------------------------
// Kernel A: mask_scores[b,n] = scores[b,n] * pred_maskiou[b,n,label[b,n]]
// ---------------------------------------------------------------------------
__global__ void mi_score_kernel(const float* __restrict__ pred_maskiou,
                                const float* __restrict__ scores,
                                const int*   __restrict__ labels,
                                float*       __restrict__ mask_scores,
                                int total) {
    int t = blockIdx.x * blockDim.x + threadIdx.x;
    if (t >= total) return;
    int lab = labels[t];
    float mi = pred_maskiou[(size_t)t * NCLS + lab];
    mask_scores[t] = scores[t] * mi;
}

// ---------------------------------------------------------------------------
// Kernel B: per-image greedy NMS (class-offset trick), LDS-resident.
//   - bitonic sort (key = valid ? mask_score : -inf, descending) in LDS
//   - async gather of boxes in sorted order into LDS
//     (global_load_async_to_lds_b128 + s_wait_asynccnt)
//   - greedy suppression over a 2048-bit LDS keep mask; barrier only after
//     alive heads (block-uniform branch), so barrier count ~= #kept boxes
//   - scatter keep (pre-cap) back to original order as 0/1 floats
// ---------------------------------------------------------------------------
__global__ void __launch_bounds__(THREADS)
mi_nms_kernel(const float* __restrict__ boxes,        // [B,N,4]
              const int*   __restrict__ labels,       // [B,N]
              const float* __restrict__ mask_scores,  // [B,N] (d_out seg0)
              float*       __restrict__ keep_out) {   // [B,N] (d_out seg2)
    __shared__ float    skey[NBOX];
    __shared__ int      sidx[NBOX];
    __shared__ float    sbox[NBOX * 4];
    __shared__ unsigned keepm[NBOX / 32];

    const int b   = blockIdx.x;
    const int tid = threadIdx.x;
    const float* bboxes = boxes       + (size_t)b * NBOX * 4;
    const int*   blab   = labels      + (size_t)b * NBOX;
    const float* bms    = mask_scores + (size_t)b * NBOX;

    __builtin_prefetch(bms + tid, 0, 0);  // global_prefetch_b8

    // --- init keys / permutation / keep mask ---
    for (int t = tid; t < NBOX; t += THREADS) {
        float ms  = bms[t];
        int   lab = blab[t];
        bool valid = (ms > SCORE_THRESH) && (lab > 0);
        skey[t] = valid ? ms : -__builtin_inff();
        sidx[t] = t;
    }
    for (int t = tid; t < NBOX / 32; t += THREADS) keepm[t] = 0xFFFFFFFFu;
    __syncthreads();

    // --- bitonic sort, descending by key, carrying index ---
    for (unsigned k = 2; k <= NBOX; k <<= 1) {
        for (unsigned j = k >> 1; j > 0; j >>= 1) {
            for (unsigned t = tid; t < NBOX; t += THREADS) {
                unsigned ixj = t ^ j;
                if (ixj > t) {
                    float a = skey[t], c = skey[ixj];
                    bool desc = ((t & k) == 0);
                    if (desc ? (a < c) : (a > c)) {
                        skey[t] = c;  skey[ixj] = a;
                        int ia = sidx[t]; sidx[t] = sidx[ixj]; sidx[ixj] = ia;
                    }
                }
            }
            __syncthreads();
        }
    }

    // --- async gather: boxes in sorted order -> LDS (per-lane gather) ---
    for (int t = tid; t < NBOX; t += THREADS) {
        const float* src = bboxes + (size_t)sidx[t] * 4;
        unsigned dst = (unsigned)(unsigned long long)(&sbox[t * 4]);
        asm volatile("global_load_async_to_lds_b128 %0, %1, off"
                     :: "v"(dst), "v"(src) : "memory");
    }
    asm volatile("s_wait_asynccnt 0x0" ::: "memory");
    __syncthreads();

    // --- apply class offset (per-class NMS trick) ---
    for (int t = tid; t < NBOX; t += THREADS) {
        float off = (float)blab[sidx[t]] * CLASS_OFFSET;
        sbox[t * 4 + 0] += off;
        sbox[t * 4 + 1] += off;
        sbox[t * 4 + 2] += off;
        sbox[t * 4 + 3] += off;
    }
    __syncthreads();

    // --- greedy suppression (matches reference fori_loop semantics) ---
    // Barrier only after alive heads: dead heads write nothing, so the last
    // barrier still fences every prior write.  Branch is block-uniform (all
    // threads read the same LDS word), so __syncthreads inside it is legal.
    for (int i = 0; i < NBOX; ++i) {
        if ((keepm[i >> 5] >> (i & 31)) & 1u) {     // uniform read
            float x1 = sbox[i * 4 + 0], y1 = sbox[i * 4 + 1];
            float x2 = sbox[i * 4 + 2], y2 = sbox[i * 4 + 3];
            float areai = (x2 - x1) * (y2 - y1);
            for (int jj = i + 1 + tid; jj < NBOX; jj += THREADS) {
                float bx1 = sbox[jj * 4 + 0], by1 = sbox[jj * 4 + 1];
                float bx2 = sbox[jj * 4 + 2], by2 = sbox[jj * 4 + 3];
                float areaj = (bx2 - bx1) * (by2 - by1);
                float iw = fmaxf(fminf(x2, bx2) - fmaxf(x1, bx1), 0.0f);
                float ih = fmaxf(fminf(y2, by2) - fmaxf(y1, by1), 0.0f);
                float inter = iw * ih;
                float iou = inter / (areai + areaj - inter + 1e-9f);
                if (iou > NMS_THRESH)
                    atomicAnd(&keepm[jj >> 5], ~(1u << (jj & 31)));
            }
            __syncthreads();
        }
    }
    __syncthreads();

    // --- scatter keep (pre-cap) back to original order; fold in `valid` ---
    for (int t = tid; t < NBOX; t += THREADS) {
        bool kept = ((keepm[t >> 5] >> (t & 31)) & 1u) &&
                    (skey[t] != -__builtin_inff());
        keep_out[(size_t)b * NBOX + sidx[t]] = kept ? 1.0f : 0.0f;
    }
}

// ---------------------------------------------------------------------------
// Kernel C: cap at DET_PER_IMG via 100th-largest original score among kept,
// then emit final_scores and final keep.
// ---------------------------------------------------------------------------
__global__ void __launch_bounds__(THREADS)
mi_cap_kernel(const float* __restrict__ scores,       // [B,N]
              const float* __restrict__ mask_scores,  // [B,N] (d_out seg0)
              float*       __restrict__ final_scores, // [B,N] (d_out seg1)
              float*       __restrict__ keep_inout) { // [B,N] (d_out seg2)
    __shared__ float skey[NBOX];
    __shared__ int   cnt;

    const int b   = blockIdx.x;
    const int tid = threadIdx.x;
    if (tid == 0) cnt = 0;
    __syncthreads();

    const float* bs   = scores       + (size_t)b * NBOX;
    const float* bms  = mask_scores  + (size_t)b * NBOX;
    float* bfin  = final_scores + (size_t)b * NBOX;
    float* bkeep = keep_inout   + (size_t)b * NBOX;

    float myks[2];
    bool  mykeep[2];
    int local = 0;
    for (int u = 0; u < 2; ++u) {
        int t = tid + u * THREADS;
        bool kp  = (bkeep[t] != 0.0f);
        float ks = kp ? bs[t] : -__builtin_inff();
        skey[t]  = ks;
        myks[u]  = ks;
        mykeep[u] = kp;
        local += kp ? 1 : 0;
    }
    atomicAdd(&cnt, local);
    __syncthreads();

    float thr = -__builtin_inff();
    if (cnt > DET_PER_IMG) {                  // uniform branch
        // bitonic sort descending (values only)
        for (unsigned k = 2; k <= NBOX; k <<= 1) {
            for (unsigned j = k >> 1; j > 0; j >>= 1) {
                for (unsigned t = tid; t < NBOX; t += THREADS) {
                    unsigned ixj = t ^ j;
                    if (ixj > t) {
                        float a = skey[t], c = skey[ixj];
                        bool desc = ((t & k) == 0);
                        if (desc ? (a < c) : (a > c)) { skey[t] = c; skey[ixj] = a; }
                    }
                }
                __syncthreads();
            }
        }
        thr = skey[DET_PER_IMG - 1];          // 100th largest
    }

    for (int u = 0; u < 2; ++u) {
        int t = tid + u * THREADS;
        bool fk = mykeep[u] && (myks[u] >= thr);
        bfin[t]  = fk ? bms[t] : 0.0f;
        bkeep[t] = fk ? 1.0f : 0.0f;
    }
}

// ---------------------------------------------------------------------------
extern "C" void kernel_launch(void* const* d_in, const int* in_sizes, int n_in,
                              void* d_out, int out_size, void* d_ws, size_t ws_size,
                              hipStream_t stream) {
    (void)in_sizes; (void)n_in; (void)out_size; (void)d_ws; (void)ws_size;

    const float* boxes        = (const float*)d_in[0];  // [B,N,4]
    const float* pred_maskiou = (const float*)d_in[1];  // [B,N,C]
    const float* scores       = (const float*)d_in[2];  // [B,N]
    const int*   labels       = (const int*)d_in[3];    // [B,N,1]

    float* out = (float*)d_out;
    float* mask_scores  = out;                    // seg0 [B,N]
    float* final_scores = out + NIMG * NBOX;      // seg1 [B,N]
    float* keep         = out + 2 * NIMG * NBOX;  // seg2 [B,N]

    const int total = NIMG * NBOX;

    mi_score_kernel<<<(total + 255) / 256, 256, 0, stream>>>(
        pred_maskiou, scores, labels, mask_scores, total);

    mi_nms_kernel<<<NIMG, THREADS, 0, stream>>>(
        boxes, labels, mask_scores, keep);

    mi_cap_kernel<<<NIMG, THREADS, 0, stream>>>(
        scores, mask_scores, final_scores, keep);
}